// AttentionCropLayer_89687507075448
// MI455X (gfx1250) — compile-verified
//
#include <hip/hip_runtime.h>
#include <cstdint>

namespace {
constexpr int TLH  = 44;           // half-width
constexpr int CROP = 88;           // window size
constexpr int INSZ = 224;          // input H/W
constexpr int CH   = 16;           // channels
constexpr int TILE = CROP * CROP;  // 7744 elements = 30976 B
constexpr int BLK  = 256;          // 8 waves (wave32)
constexpr int ROW4 = CROP / 4;     // 22 float4 per row
constexpr int TIL4 = TILE / 4;     // 1936 float4 per tile
}

typedef unsigned int u32v4 __attribute__((ext_vector_type(4)));
typedef int          i32v8 __attribute__((ext_vector_type(8)));
typedef int          i32v4 __attribute__((ext_vector_type(4)));

// One block per (batch, channel) plane: TDM DMA the 88x88 window into LDS,
// then 8 waves apply the separable sigmoid mask and store b128-coalesced.
__global__ __launch_bounds__(BLK)
void attn_crop_tdm(const float* __restrict__ images,
                   const float* __restrict__ locs,
                   float* __restrict__ out) {
  __shared__ alignas(16) float tile[TILE];
  __shared__ alignas(16) float prof[CROP];

  const int bc = (int)blockIdx.x;   // plane index = b*16 + c
  const int b  = bc >> 4;

  // --- separable soft-mask profile (identical for every block; recompute) ---
  if (threadIdx.x < CROP) {
    const float r  = (float)threadIdx.x;
    const float s0 = 1.0f / (1.0f + expf(-10.0f * r));
    const float s1 = 1.0f / (1.0f + expf(-10.0f * (r - (float)CROP)));
    prof[threadIdx.x] = s0 - s1;
  }

  // --- window offsets (uniform per block; force into SGPRs) ---
  const float lx = locs[2 * b + 0];
  const float ly = locs[2 * b + 1];
  const float tx = fminf(fmaxf(lx * 224.0f, (float)TLH), (float)(INSZ - TLH));
  const float ty = fminf(fmaxf(ly * 224.0f, (float)TLH), (float)(INSZ - TLH));
  const int w_off = __builtin_amdgcn_readfirstlane((int)floorf(tx - (float)TLH));
  const int h_off = __builtin_amdgcn_readfirstlane((int)floorf(ty - (float)TLH));

  const float* src =
      images + ((size_t)bc * (INSZ * INSZ) + (size_t)w_off * INSZ + (size_t)h_off);

#if __has_builtin(__builtin_amdgcn_tensor_load_to_lds)
  // -------- Tensor Data Mover path: one 2D-tile DMA per block --------
  if (threadIdx.x < 32) {  // single wave issues the TDM op (EXEC ignored by TDM)
    const uint64_t gaddr  = (uint64_t)(uintptr_t)src;
    const uint32_t ldsoff = (uint32_t)(uintptr_t)(void*)&tile[0];  // LDS byte offset

    // D# group 0 (128b): count=1 valid, lds_addr, global_addr[56:0], type=2
    u32v4 g0;
    g0.x = 1u;                                                    // count=1, user mode
    g0.y = ldsoff;                                                // lds_addr (bytes)
    g0.z = (uint32_t)gaddr;                                       // global_addr[31:0]
    g0.w = (uint32_t)((gaddr >> 32) & 0x1FFFFFFull) | (2u << 30); // addr[56:32] | type=2

    // D# group 1 (256b): data_size=4B, tensor dims/strides, tile dims
    i32v8 g1;
    g1[0] = (int)(2u << 16);                 // data_size = 2 -> 4 bytes; no multicast
    g1[1] = (int)((unsigned)INSZ << 16);     // tensor_dim0[15:0] = 224 (bits 63:48)
    g1[2] = (int)((unsigned)INSZ << 16);     // tensor_dim0 hi=0 | tensor_dim1[15:0]=224
    g1[3] = (int)((unsigned)CROP << 16);     // tensor_dim1 hi=0 | tile_dim0 = 88
    g1[4] = CROP;                            // tile_dim1 = 88, tile_dim2 = 0
    g1[5] = INSZ;                            // tensor_dim0_stride = 224 elements
    g1[6] = (int)(((unsigned)(INSZ * INSZ) & 0xFFFFu) << 16); // dim1_stride lo (unused 2D)
    g1[7] = (int)((unsigned)(INSZ * INSZ) >> 16);             // dim1_stride hi

    i32v4 z4 = {0, 0, 0, 0};                 // groups 2/3: 2D tensor, unused
    i32v8 z8 = {0, 0, 0, 0, 0, 0, 0, 0};     // trailing group (clang-23 6-arg form)
    __builtin_amdgcn_tensor_load_to_lds(g0, g1, z4, z4, z8, 0);

#if __has_builtin(__builtin_amdgcn_s_wait_tensorcnt)
    __builtin_amdgcn_s_wait_tensorcnt(0);
#else
    asm volatile("s_wait_tensorcnt 0" ::: "memory");
#endif
  }
#else
  // -------- Fallback: cooperative coalesced loads into LDS --------
  for (int e = (int)threadIdx.x; e < TILE; e += BLK) {
    const int i = e / CROP;
    const int j = e - i * CROP;
    tile[e] = src[i * INSZ + j];
  }
#endif

  __syncthreads();  // tile visible to all 8 waves

  // --- apply separable mask; float4 never spans a row (4 | 88) ---
  const float4* tile4 = reinterpret_cast<const float4*>(tile);
  const float4* prof4 = reinterpret_cast<const float4*>(prof);
  float4* dst4 = reinterpret_cast<float4*>(out + (size_t)bc * TILE);

  for (int v = (int)threadIdx.x; v < TIL4; v += BLK) {
    const int i  = v / ROW4;          // row index 0..87
    const int q  = v - i * ROW4;      // float4 index within row, 0..21
    const float  pi = prof[i];        // ds_load_b32 (broadcast)
    const float4 pj = prof4[q];       // ds_load_b128 (16B-aligned)
    const float4 t  = tile4[v];       // ds_load_b128

    float4 r;
    r.x = t.x * pi * pj.x;
    r.y = t.y * pi * pj.y;
    r.z = t.z * pi * pj.z;
    r.w = t.w * pi * pj.w;
    dst4[v] = r;                      // global_store_b128, coalesced
  }
}

extern "C" void kernel_launch(void* const* d_in, const int* in_sizes, int n_in,
                              void* d_out, int out_size, void* d_ws, size_t ws_size,
                              hipStream_t stream) {
  const float* images = (const float*)d_in[0];   // [B,16,224,224] f32
  const float* locs   = (const float*)d_in[1];   // [B,2] f32
  float* out = (float*)d_out;                    // [B,16,88,88] f32

  const int B = in_sizes[1] / 2;                 // 128
  const int nblocks = B * CH;                    // 2048 planes
  attn_crop_tdm<<<nblocks, BLK, 0, stream>>>(images, locs, out);
}